// FourierNeuralOperator_45346264711542
// MI455X (gfx1250) — compile-verified
//
#include <hip/hip_runtime.h>
#include <math.h>

// Problem constants (fixed by the reference)
#define Bn     32
#define Sn     8192
#define Hn     128
#define Ln     4
#define MODESn 16
#define SCHUNKS 32
#define KC (Sn / SCHUNKS)   // 256 s-values per forward-DFT block

typedef float v2f __attribute__((ext_vector_type(2)));
typedef float v8f __attribute__((ext_vector_type(8)));

static constexpr float TWO_PI_OVER_S = 6.28318530717958647692f / (float)Sn;

// D = A(16x4 f32) * B(4x16 f32) + C(16x16 f32), full-precision WMMA.
__device__ __forceinline__ v8f wmma4(v2f a, v2f b, v8f c) {
  return __builtin_amdgcn_wmma_f32_16x16x4_f32(false, a, false, b, (short)0, c,
                                               false, false);
}

// ---------------------------------------------------------------------------
// 16-mode DFT of the raw input signal:
//   X_m[b] = sum_s x[b,s] * e^{-2*pi*i*m*s/S},  m = 0..15
// One block per (m, b); 256 threads partial-sum with a phasor recurrence,
// then LDS tree reduction. Tiny kernel (1 MiB of x read 16x).
// ---------------------------------------------------------------------------
__global__ void __launch_bounds__(256)
dft_x_kernel(const float* __restrict__ x, float* __restrict__ XRe,
             float* __restrict__ XIm) {
  const int b = blockIdx.y;
  const int m = blockIdx.x;   // 0..15
  const int t = threadIdx.x;  // 0..255
  __shared__ float rRe[256], rIm[256];

  float sv, cv;
  __sincosf((float)((m * t) & (Sn - 1)) * TWO_PI_OVER_S, &sv, &cv);
  float c = cv, ns = -sv;  // e^{-i theta} = (cos, -sin)
  __sincosf((float)((m * 256) & (Sn - 1)) * TWO_PI_OVER_S, &sv, &cv);
  const float rc = cv, rns = -sv;  // rotation by 256 samples

  float ar = 0.f, ai = 0.f;
  const float* xb = x + (size_t)b * Sn;
  for (int s = t; s < Sn; s += 256) {
    const float xv = xb[s];
    ar = fmaf(xv, c, ar);
    ai = fmaf(xv, ns, ai);
    const float nc = c * rc - ns * rns;
    ns = fmaf(c, rns, ns * rc);
    c = nc;
  }
  rRe[t] = ar;
  rIm[t] = ai;
  __syncthreads();
  for (int off = 128; off > 0; off >>= 1) {
    if (t < off) { rRe[t] += rRe[t + off]; rIm[t] += rIm[t + off]; }
    __syncthreads();
  }
  if (t == 0) {
    XRe[b * MODESn + m] = rRe[0];
    XIm[b * MODESn + m] = rIm[0];
  }
}

// ---------------------------------------------------------------------------
// Layer-0 forward coefficients without materializing h:
//   c_m[b,k] = pw[k]*X_m[b] + S*pb[k]*[m==0]
// Written in the (nchunks==1) partials layout consumed by mix_kernel.
// ---------------------------------------------------------------------------
__global__ void __launch_bounds__(128)
coef0_kernel(const float* __restrict__ XRe, const float* __restrict__ XIm,
             const float* __restrict__ pw, const float* __restrict__ pb,
             float* __restrict__ cRe, float* __restrict__ cIm) {
  const int b = blockIdx.x >> 4;
  const int m = blockIdx.x & 15;
  const int k = threadIdx.x;
  const float xr = XRe[b * MODESn + m];
  const float xi = XIm[b * MODESn + m];
  const float w = pw[k];
  float re = w * xr;
  const float im = w * xi;
  if (m == 0) re = fmaf((float)Sn, pb[k], re);
  const size_t o = ((size_t)b * MODESn + m) * Hn + k;
  cRe[o] = re;
  cIm[o] = im;
}

// ---------------------------------------------------------------------------
// Forward partial DFT (layers >= 1): per (batch, s-chunk) block,
//   pRe/pIm[chunk,b,m,k] = sum_{s in chunk} h[b,s,k] * e^{-2*pi*i*m*s/S}
// Each block = 4 waves; wave w owns channels [32w, 32w+32).
// WMMA tiles: A = twiddle 16x4 (modes x s), B = h 4x16 (s x channels).
// Twiddles advance by a constant rotation per k-step -> 8 FMAs/iter,
// no transcendental ops in the hot loop (keeps the loop WMMA-limited).
// ---------------------------------------------------------------------------
__global__ void __launch_bounds__(128)
fwd_dft_kernel(const float* __restrict__ h, float* __restrict__ pRe,
               float* __restrict__ pIm) {
  const int b     = blockIdx.y;
  const int chunk = blockIdx.x;
  const int wave  = threadIdx.x >> 5;
  const int lane  = threadIdx.x & 31;
  const int m     = lane & 15;          // A-matrix row (mode)
  const int n     = lane & 15;          // B/C-matrix column (channel)
  const int khalf = (lane >> 4) << 1;   // 0 or 2: K-half per ISA layout
  const int nb    = wave * 32;
  const int s0    = chunk * KC;
  const float* hb = h + (size_t)b * Sn * Hn;

  // Initial phasors e^{-i*2pi*m*s/S} at s = s0+khalf and s0+khalf+1,
  // plus the per-iteration rotation e^{-i*2pi*4m/S}.
  float c0, ns0, c1, ns1, rc, rns;
  {
    float sv, cv;
    __sincosf((float)((m * (s0 + khalf)) & (Sn - 1)) * TWO_PI_OVER_S, &sv, &cv);
    c0 = cv; ns0 = -sv;
    __sincosf((float)((m * (s0 + khalf + 1)) & (Sn - 1)) * TWO_PI_OVER_S, &sv, &cv);
    c1 = cv; ns1 = -sv;
    __sincosf((float)((4 * m) & (Sn - 1)) * TWO_PI_OVER_S, &sv, &cv);
    rc = cv; rns = -sv;
  }

  v8f aC0 = {}; v8f aC1 = {}; v8f aS0 = {}; v8f aS1 = {};

  for (int k0 = 0; k0 < KC; k0 += 4) {
    const int s = s0 + k0 + khalf;
    v2f Ac = {c0, c1};
    v2f As = {ns0, ns1};
    const float* p = hb + (size_t)s * Hn;
    v2f B0 = {p[nb + n],      p[Hn + nb + n]};
    v2f B1 = {p[nb + 16 + n], p[Hn + nb + 16 + n]};
    aC0 = wmma4(Ac, B0, aC0);
    aC1 = wmma4(Ac, B1, aC1);
    aS0 = wmma4(As, B0, aS0);
    aS1 = wmma4(As, B1, aS1);
    // Rotate both phasors by e^{-i*2pi*4m/S}.
    const float t0 = c0 * rc - ns0 * rns;
    ns0 = fmaf(c0, rns, ns0 * rc);
    c0 = t0;
    const float t1 = c1 * rc - ns1 * rns;
    ns1 = fmaf(c1, rns, ns1 * rc);
    c1 = t1;
  }

  // C/D layout: VGPR r -> row r (lanes 0-15) / row r+8 (lanes 16-31).
  const int mrow = (lane >> 4) * 8;
  float* oRe = pRe + (((size_t)chunk * Bn + b) * MODESn) * Hn;
  float* oIm = pIm + (((size_t)chunk * Bn + b) * MODESn) * Hn;
#pragma unroll
  for (int r = 0; r < 8; ++r) {
    const int mm = mrow + r;
    oRe[mm * Hn + nb + n]      = aC0[r];
    oRe[mm * Hn + nb + 16 + n] = aC1[r];
    oIm[mm * Hn + nb + n]      = aS0[r];
    oIm[mm * Hn + nb + 16 + n] = aS1[r];
  }
}

// ---------------------------------------------------------------------------
// Mode mixing: reduce chunk partials, then for each (b,m):
//   d[k] = sum_h c[b,m,h] * W[h,k,m]   (complex)
// Fold the irfft scaling into E:
//   E[b, m,    k] = Re(d)*(m==0 ? 1/S : 2/S)      (cos coefficients)
//   E[b, 16+m, k] = (m==0 ? 0 : -Im(d)*2/S)       (sin coefficients)
// (irfft's C2R drops Im of the DC bin, matching jnp.fft.irfft.)
// ---------------------------------------------------------------------------
__global__ void __launch_bounds__(128)
mix_kernel(const float* __restrict__ pRe, const float* __restrict__ pIm,
           const float* __restrict__ wRe, const float* __restrict__ wIm,
           float* __restrict__ E, int nchunks) {
  const int b = blockIdx.x >> 4;
  const int m = blockIdx.x & 15;
  const int t = threadIdx.x;  // 0..127: used as h for reduction, then as k
  __shared__ float sRe[Hn], sIm[Hn];

  float accR = 0.f, accI = 0.f;
  const size_t base = ((size_t)b * MODESn + m) * Hn + t;
  for (int ch = 0; ch < nchunks; ++ch) {
    const size_t off = (size_t)ch * Bn * MODESn * Hn + base;
    accR += pRe[off];
    accI += pIm[off];
  }
  sRe[t] = accR;
  sIm[t] = accI;
  __syncthreads();

  float dr = 0.f, di = 0.f;
  const int k = t;
  for (int hh = 0; hh < Hn; ++hh) {
    const float cr = sRe[hh], ci = sIm[hh];
    const size_t wo = ((size_t)hh * Hn + k) * MODESn + m;
    const float wr = wRe[wo], wi = wIm[wo];
    dr = fmaf(cr, wr, dr);
    dr = fmaf(-ci, wi, dr);
    di = fmaf(cr, wi, di);
    di = fmaf(ci, wr, di);
  }
  const float inv = 1.0f / (float)Sn;
  E[((size_t)b * 32 + m) * Hn + k]      = dr * ((m == 0) ? inv : 2.0f * inv);
  E[((size_t)b * 32 + 16 + m) * Hn + k] = (m == 0) ? 0.0f : -2.0f * inv * di;
}

// ---------------------------------------------------------------------------
// Inverse partial DFT + exact GELU:
//   h[b,s,k] = gelu( sum_{j<16} E[b,j,k]*cos(2pi j s/S)
//                  + sum_{j<16} E[b,16+j,k]*sin(2pi j s/S) )
// Per block: 64 s-rows (wave -> 16-row M tile), E_b staged in LDS (16 KB).
// 16 sincos per lane amortized over 64 WMMAs.
// ---------------------------------------------------------------------------
__global__ void __launch_bounds__(128)
inv_dft_gelu_kernel(const float* __restrict__ E, float* __restrict__ h) {
  const int b     = blockIdx.y;
  const int stile = blockIdx.x;  // 0..S/64-1
  const int wave  = threadIdx.x >> 5;
  const int lane  = threadIdx.x & 31;
  __shared__ float Es[32 * Hn];  // 4096 floats = 16 KB

  const float* Eb = E + (size_t)b * 32 * Hn;
  for (int i = threadIdx.x; i < 32 * Hn / 4; i += 128)
    ((float4*)Es)[i] = ((const float4*)Eb)[i];
  __syncthreads();

  const int sbase = stile * 64 + wave * 16;
  const int khalf = (lane >> 4) << 1;
  const int s     = sbase + (lane & 15);  // A-matrix row = s within tile
  const int n     = lane & 15;            // B/C column

  // Precompute all 8 A fragments (K = 32 total: cols 0-15 cos, 16-31 sin).
  v2f a[8];
#pragma unroll
  for (int t = 0; t < 8; ++t) {
#pragma unroll
    for (int u = 0; u < 2; ++u) {
      const int j  = 4 * t + khalf + u;
      const int mm = j & 15;
      float sv, cv;
      __sincosf((float)((mm * s) & (Sn - 1)) * TWO_PI_OVER_S, &sv, &cv);
      a[t][u] = (t < 4) ? cv : sv;  // t<4 <=> j<16 (wave-uniform)
    }
  }

  float* hb = h + ((size_t)b * Sn + sbase) * Hn;
#pragma unroll
  for (int nt = 0; nt < 8; ++nt) {
    const int nbase = nt * 16;
    v8f acc = {};
#pragma unroll
    for (int t = 0; t < 8; ++t) {
      v2f Bf = {Es[(4 * t + khalf) * Hn + nbase + n],
                Es[(4 * t + khalf + 1) * Hn + nbase + n]};
      acc = wmma4(a[t], Bf, acc);
    }
    const int mrow = (lane >> 4) * 8;
#pragma unroll
    for (int r = 0; r < 8; ++r) {
      const float v = acc[r];
      const float g = 0.5f * v * (1.0f + erff(v * 0.70710678118654752440f));
      hb[(size_t)(mrow + r) * Hn + nbase + n] = g;
    }
  }
}

// ---------------------------------------------------------------------------
// out[b,s] = h[b,s,:] . out_w + out_b
// ---------------------------------------------------------------------------
__global__ void __launch_bounds__(256)
out_kernel(const float* __restrict__ h, const float* __restrict__ ow,
           const float* __restrict__ ob, float* __restrict__ out) {
  const int t = blockIdx.x * blockDim.x + threadIdx.x;  // over B*S
  if (t >= Bn * Sn) return;
  const float4* hp = (const float4*)(h + (size_t)t * Hn);
  const float4* wp = (const float4*)ow;
  float acc = 0.f;
#pragma unroll
  for (int i = 0; i < Hn / 4; ++i) {
    const float4 hv = hp[i];
    const float4 wv = wp[i];
    acc = fmaf(hv.x, wv.x, acc);
    acc = fmaf(hv.y, wv.y, acc);
    acc = fmaf(hv.z, wv.z, acc);
    acc = fmaf(hv.w, wv.w, acc);
  }
  out[t] = acc + ob[0];
}

// ---------------------------------------------------------------------------
extern "C" void kernel_launch(void* const* d_in, const int* in_sizes, int n_in,
                              void* d_out, int out_size, void* d_ws,
                              size_t ws_size, hipStream_t stream) {
  (void)in_sizes; (void)n_in; (void)out_size;
  const float* x   = (const float*)d_in[0];
  const float* pw  = (const float*)d_in[1];
  const float* pb  = (const float*)d_in[2];
  const float* wRe = (const float*)d_in[3];
  const float* wIm = (const float*)d_in[4];
  const float* ow  = (const float*)d_in[5];
  const float* ob  = (const float*)d_in[6];
  float* out = (float*)d_out;

  const size_t hBytes = (size_t)Bn * Sn * Hn * sizeof(float);               // 128 MiB
  const size_t pBytes = (size_t)SCHUNKS * Bn * MODESn * Hn * sizeof(float); // 8 MiB
  const size_t eBytes = (size_t)Bn * 32 * Hn * sizeof(float);               // 0.5 MiB
  const size_t xBytes = (size_t)Bn * MODESn * sizeof(float);                // 2 KiB
  if (ws_size < hBytes + 2 * pBytes + eBytes + 2 * xBytes) return;  // defensive

  char* ws = (char*)d_ws;
  float* h   = (float*)ws;                                   ws += hBytes;
  float* pRe = (float*)ws;                                   ws += pBytes;
  float* pIm = (float*)ws;                                   ws += pBytes;
  float* E   = (float*)ws;                                   ws += eBytes;
  float* XRe = (float*)ws;                                   ws += xBytes;
  float* XIm = (float*)ws;

  // Layer 0: analytic forward transform of the affine projection.
  //   c_m = pw * DFT_16(x)_m + S*pb*[m==0]  (no 128 MiB h round-trip)
  dft_x_kernel<<<dim3(MODESn, Bn), 256, 0, stream>>>(x, XRe, XIm);
  coef0_kernel<<<Bn * MODESn, 128, 0, stream>>>(XRe, XIm, pw, pb, pRe, pIm);
  mix_kernel<<<Bn * MODESn, 128, 0, stream>>>(pRe, pIm, wRe, wIm, E, 1);
  inv_dft_gelu_kernel<<<dim3(Sn / 64, Bn), 128, 0, stream>>>(E, h);

  // Layers 1..3: generic WMMA DFT path on h.
  for (int l = 1; l < Ln; ++l) {
    const size_t woff = (size_t)l * Hn * Hn * MODESn;
    fwd_dft_kernel<<<dim3(SCHUNKS, Bn), 128, 0, stream>>>(h, pRe, pIm);
    mix_kernel<<<Bn * MODESn, 128, 0, stream>>>(pRe, pIm, wRe + woff,
                                                wIm + woff, E, SCHUNKS);
    inv_dft_gelu_kernel<<<dim3(Sn / 64, Bn), 128, 0, stream>>>(E, h);
  }

  out_kernel<<<(Bn * Sn) / 256, 256, 0, stream>>>(h, ow, ob, out);
}